// MultiHeadCrossAttention_78503412236770
// MI455X (gfx1250) — compile-verified
//
#include <hip/hip_runtime.h>

typedef float v2f __attribute__((ext_vector_type(2)));
typedef float v8f __attribute__((ext_vector_type(8)));

#define B_    128
#define DX_   1024
#define NK_   1024
#define DKIN_ 512
#define H_    16
#define DK_   64
#define HDK_  1024
#define DO_   512

// ---------------------------------------------------------------------------
// attn_out is exactly all-ones (softmax over a singleton axis).
__global__ void fill_ones_kernel(float4* __restrict__ out) {
    size_t i = (size_t)blockIdx.x * blockDim.x + threadIdx.x;
    out[i] = make_float4(1.f, 1.f, 1.f, 1.f);
}

// ---------------------------------------------------------------------------
// Stage 1 of ksum[b,d] = sum_n k[b,n,d]: each block sums a 128-row chunk.
// grid = (B_*8), block = 128 threads, thread owns one float4 column.
__global__ void reduce_k_partial(const float* __restrict__ k, float* __restrict__ P) {
    int b = blockIdx.x >> 3;
    int s = blockIdx.x & 7;
    int t = threadIdx.x;                       // float4 column 0..127
    const float4* src = (const float4*)(k + (size_t)b * NK_ * DKIN_ + (size_t)s * 128 * DKIN_);
    float4 acc = make_float4(0.f, 0.f, 0.f, 0.f);
#pragma unroll 8
    for (int n = 0; n < 128; ++n) {
        float4 v = src[(size_t)n * (DKIN_ / 4) + t];
        acc.x += v.x; acc.y += v.y; acc.z += v.z; acc.w += v.w;
    }
    ((float4*)P)[(size_t)blockIdx.x * 128 + t] = acc;
}

// Stage 2: fold the 8 partials. grid = B_, block = 128.
__global__ void reduce_k_final(const float* __restrict__ P, float* __restrict__ ksum) {
    int b = blockIdx.x, t = threadIdx.x;
    float4 acc = make_float4(0.f, 0.f, 0.f, 0.f);
#pragma unroll
    for (int s = 0; s < 8; ++s) {
        float4 v = ((const float4*)P)[(size_t)(b * 8 + s) * 128 + t];
        acc.x += v.x; acc.y += v.y; acc.z += v.z; acc.w += v.w;
    }
    ((float4*)ksum)[(size_t)b * 128 + t] = acc;
}

// ---------------------------------------------------------------------------
// BN1: biased batch stats per column of x (B_=128 rows). grid=8, block=128.
__global__ void bn1_stats(const float* __restrict__ x, const float* __restrict__ g,
                          const float* __restrict__ bb, float* __restrict__ sc,
                          float* __restrict__ sh) {
    int col = blockIdx.x * 128 + threadIdx.x;
    float s = 0.f, s2 = 0.f;
#pragma unroll 4
    for (int r = 0; r < B_; ++r) {
        float v = x[(size_t)r * DX_ + col];
        s += v; s2 += v * v;
    }
    float mean = s * (1.f / B_);
    float var  = s2 * (1.f / B_) - mean * mean;
    float rstd = rsqrtf(var + 1e-5f);
    float scale = g[col] * rstd;
    sc[col] = scale;
    sh[col] = bb[col] - mean * scale;
}

// xhat = x*scale + shift. grid=B_, block=256 (4 cols per thread via float4).
__global__ void apply_bn1(const float* __restrict__ x, const float* __restrict__ sc,
                          const float* __restrict__ sh, float* __restrict__ xhat) {
    int r = blockIdx.x, c = threadIdx.x;
    float4 xv = ((const float4*)(x + (size_t)r * DX_))[c];
    float4 s  = ((const float4*)sc)[c];
    float4 h  = ((const float4*)sh)[c];
    float4 o;
    o.x = xv.x * s.x + h.x; o.y = xv.y * s.y + h.y;
    o.z = xv.z * s.z + h.z; o.w = xv.w * s.w + h.w;
    ((float4*)(xhat + (size_t)r * DX_))[c] = o;
}

// BN2 stats over the (H*B, DK) view: column d gathers xq[b, h*DK+d] for all h,b.
// grid=DK_, block=128, deterministic LDS tree reduction.
__global__ void bn2_stats(const float* __restrict__ xq, const float* __restrict__ g,
                          const float* __restrict__ bb, float* __restrict__ sc,
                          float* __restrict__ sh) {
    __shared__ float ss[128], ss2[128];
    int d = blockIdx.x, t = threadIdx.x;
    float s = 0.f, s2 = 0.f;
    for (int j = t; j < H_ * B_; j += 128) {
        int b = j & (B_ - 1);
        int h = j >> 7;
        float v = xq[(size_t)b * HDK_ + h * DK_ + d];
        s += v; s2 += v * v;
    }
    ss[t] = s; ss2[t] = s2;
    __syncthreads();
    for (int off = 64; off > 0; off >>= 1) {
        if (t < off) { ss[t] += ss[t + off]; ss2[t] += ss2[t + off]; }
        __syncthreads();
    }
    if (t == 0) {
        float mean = ss[0] * (1.f / (H_ * B_));
        float var  = ss2[0] * (1.f / (H_ * B_)) - mean * mean;
        float rstd = rsqrtf(var + 1e-5f);
        float scale = g[d] * rstd;
        sc[d] = scale;
        sh[d] = bb[d] - mean * scale;
    }
}

// ---------------------------------------------------------------------------
// Generic fp32 WMMA GEMM: C[m,n] = sum_k A[m,k]*W[n,k] + biasScale*bias[n].
// One wave per 16x16 tile, V_WMMA_F32_16X16X4_F32, K stepped by 4.
// ISA layouts: A/B frag = 2 consecutive K at koff=2*(lane>=16); C/D: M=v+8*(lane>=16), N=lane&15.
__global__ void __launch_bounds__(256)
gemm_wmma_f32(const float* __restrict__ A, int lda,
              const float* __restrict__ W, int ldw,
              const float* __restrict__ bias, float biasScale,
              float* __restrict__ C, int ldc,
              int tilesN, int K) {
    int lane = threadIdx.x & 31;
    int tile = blockIdx.x * 8 + (threadIdx.x >> 5);
    int tm = (tile / tilesN) * 16;
    int tn = (tile % tilesN) * 16;
    int half = lane >> 4;
    int l15  = lane & 15;
    int m = tm + l15;
    int n = tn + l15;
    int koff = half * 2;
    const float* arow = A + (size_t)m * lda + koff;
    const float* wrow = W + (size_t)n * ldw + koff;
    v8f acc = {};
#pragma unroll 8
    for (int kk = 0; kk < K; kk += 4) {
        v2f a = *(const v2f*)(arow + kk);
        v2f b = *(const v2f*)(wrow + kk);
        acc = __builtin_amdgcn_wmma_f32_16x16x4_f32(false, a, false, b,
                                                    (short)0, acc, false, false);
    }
    float bz = biasScale * bias[n];
#pragma unroll
    for (int v = 0; v < 8; ++v) {
        int row = tm + v + half * 8;
        C[(size_t)row * ldc + n] = acc[v] + bz;
    }
}

// latent = BN2(xq_view) @ Wlin.T + blin, stored directly permuted as
// x_hid[b, h*DK+d] = latent[h*B+b, d].  M=2048, N=K=64. grid=64, block=256.
__global__ void __launch_bounds__(256)
gemm_latent_xhid(const float* __restrict__ xq,
                 const float* __restrict__ sc2, const float* __restrict__ sh2,
                 const float* __restrict__ Wlin, const float* __restrict__ blin,
                 float* __restrict__ xhid) {
    int lane = threadIdx.x & 31;
    int tile = blockIdx.x * 8 + (threadIdx.x >> 5);
    int tm = (tile >> 2) * 16;   // tilesN = 64/16 = 4
    int tn = (tile & 3) * 16;
    int half = lane >> 4;
    int l15  = lane & 15;
    int hb = tm + l15;
    int h = hb >> 7, b = hb & (B_ - 1);
    int n = tn + l15;
    int koff = half * 2;
    const float* arow = xq + (size_t)b * HDK_ + h * DK_ + koff;
    const float* wrow = Wlin + (size_t)n * DK_ + koff;
    v8f acc = {};
#pragma unroll
    for (int kk = 0; kk < DK_; kk += 4) {
        v2f xa = *(const v2f*)(arow + kk);
        v2f s  = *(const v2f*)(sc2 + koff + kk);
        v2f sh = *(const v2f*)(sh2 + koff + kk);
        v2f a  = xa * s + sh;                 // fused BN2 on the A operand
        v2f bf = *(const v2f*)(wrow + kk);
        acc = __builtin_amdgcn_wmma_f32_16x16x4_f32(false, a, false, bf,
                                                    (short)0, acc, false, false);
    }
    float bz = blin[n];
#pragma unroll
    for (int v = 0; v < 8; ++v) {
        int hb2 = tm + v + half * 8;
        int h2 = hb2 >> 7, b2 = hb2 & (B_ - 1);
        xhid[(size_t)b2 * HDK_ + h2 * DK_ + tn + l15] = acc[v] + bz;
    }
}

// ---------------------------------------------------------------------------
extern "C" void kernel_launch(void* const* d_in, const int* in_sizes, int n_in,
                              void* d_out, int out_size, void* d_ws, size_t ws_size,
                              hipStream_t stream) {
    const float* x     = (const float*)d_in[0];
    const float* k     = (const float*)d_in[1];
    const float* bn1_g = (const float*)d_in[2];
    const float* bn1_b = (const float*)d_in[3];
    const float* Wx    = (const float*)d_in[4];
    const float* bx    = (const float*)d_in[5];
    const float* Wk    = (const float*)d_in[6];
    const float* bk    = (const float*)d_in[7];
    const float* bn2_g = (const float*)d_in[8];
    const float* bn2_b = (const float*)d_in[9];
    const float* Wlin  = (const float*)d_in[10];
    const float* blin  = (const float*)d_in[11];
    const float* Wo    = (const float*)d_in[12];
    const float* bo    = (const float*)d_in[13];

    float* out = (float*)d_out;
    float* ws  = (float*)d_ws;

    // workspace layout (float offsets)
    float* P    = ws + 0;        // 128*8*512 = 524288
    float* ksum = ws + 524288;   // 65536
    float* sc1  = ws + 589824;   // 1024
    float* sh1  = ws + 590848;   // 1024
    float* xhat = ws + 591872;   // 131072
    float* xq   = ws + 722944;   // 131072
    float* sc2  = ws + 854016;   // 64
    float* sh2  = ws + 854080;   // 64
    float* xhid = ws + 854144;   // 131072
    float* outm = ws + 985216;   // 131072  (total 1116288 floats ~= 4.26 MB)

    float* attn_out = out;                        // 128*16*1024 = 2097152
    float* out2     = out + 2097152;              // 128*512
    float* out3     = out + 2097152 + 65536;      // 128*512

    // 1) attn == softmax over singleton axis == 1.0 exactly
    fill_ones_kernel<<<2048, 256, 0, stream>>>((float4*)attn_out);

    // 2) ksum[b,:] = sum_n k[b,n,:]   (the only HBM-heavy stage: 256 MB)
    reduce_k_partial<<<B_ * 8, 128, 0, stream>>>(k, P);
    reduce_k_final<<<B_, 128, 0, stream>>>(P, ksum);

    // 3) BN1 + xq = BN1(x) @ Wx.T + bx        (M=128, N=1024, K=1024)
    bn1_stats<<<8, 128, 0, stream>>>(x, bn1_g, bn1_b, sc1, sh1);
    apply_bn1<<<B_, 256, 0, stream>>>(x, sc1, sh1, xhat);
    gemm_wmma_f32<<<64, 256, 0, stream>>>(xhat, DX_, Wx, DX_, bx, 1.0f,
                                          xq, HDK_, HDK_ / 16, DX_);

    // 4) BN2 over (H*B, DK) view; x_hid = permute(latent)
    bn2_stats<<<DK_, 128, 0, stream>>>(xq, bn2_g, bn2_b, sc2, sh2);
    gemm_latent_xhid<<<64, 256, 0, stream>>>(xq, sc2, sh2, Wlin, blin, xhid);

    // 5) out (attn==1) = ksum @ Wk.T + NK*bk  (M=128, N=1024, K=512)
    gemm_wmma_f32<<<64, 256, 0, stream>>>(ksum, DKIN_, Wk, DKIN_, bk, (float)NK_,
                                          outm, HDK_, HDK_ / 16, DKIN_);

    // 6) final projections through Wo          (M=128, N=512, K=1024)
    gemm_wmma_f32<<<32, 256, 0, stream>>>(outm, HDK_, Wo, HDK_, bo, 1.0f,
                                          out2, DO_, DO_ / 16, HDK_);
    gemm_wmma_f32<<<32, 256, 0, stream>>>(xhid, HDK_, Wo, HDK_, bo, 1.0f,
                                          out3, DO_, DO_ / 16, HDK_);
}